// Net_88502096101846
// MI455X (gfx1250) — compile-verified
//
#include <hip/hip_runtime.h>
#include <hip/hip_bf16.h>
#include <math.h>

// ---------------------------------------------------------------------------
// 2-layer GCN, MI455X (gfx1250, wave32).
//   deg[d]   = 1 + #edges into d                         (atomic f32)
//   dinv[i]  = rsqrt(deg[i]); xd[i] = x[i]*dinv[i]
//   t[d]     = sum_{e:dst=d} xd[src]                     (scalar scatter!)
//   h1[i][j] = relu(dinv[i]*(t[i]+xd[i]) * W1[j] + b1[j])   (rank-1)
//   g[i][c]  = dinv[i] * (h1[i,:] @ W2[:,c])             (WMMA f32 16x16x4)
//   u[d][c]  = sum_{e:dst=d} g[src][c]                   (2-wide scatter)
//   out[i]   = log_softmax(dinv[i]*(u[i]+g[i]) + b2)
// ---------------------------------------------------------------------------

typedef __attribute__((ext_vector_type(2))) float v2f;
typedef __attribute__((ext_vector_type(8))) float v8f;

// ---- init: deg=1 (self loop), t=0, u=0 -----------------------------------
__global__ void gcn_init(float* __restrict__ deg, float* __restrict__ t,
                         float* __restrict__ u, int N) {
  int i = blockIdx.x * blockDim.x + threadIdx.x;
  if (i < N) {
    deg[i] = 1.0f;
    t[i] = 0.0f;
    u[2 * i + 0] = 0.0f;
    u[2 * i + 1] = 0.0f;
  }
}

// ---- in-degree (edges only; self-loop preloaded as 1) --------------------
__global__ void gcn_degree(const int* __restrict__ dst, float* __restrict__ deg,
                           int E) {
  int e = blockIdx.x * blockDim.x + threadIdx.x;
  if (e < E) atomicAdd(&deg[dst[e]], 1.0f);
}

// ---- dinv + premultiplied feature ----------------------------------------
__global__ void gcn_dinv(const float* __restrict__ x, float* __restrict__ deg,
                         float* __restrict__ xd, int N) {
  int i = blockIdx.x * blockDim.x + threadIdx.x;
  if (i < N) {
    float d = deg[i];
    float di = (d > 0.0f) ? rsqrtf(d) : 0.0f;
    deg[i] = di;              // becomes dinv in place
    xd[i] = x[i] * di;
  }
}

// ---- layer-1 scalar scatter: t[dst] += xd[src] ---------------------------
__global__ void gcn_scatter1(const int* __restrict__ src,
                             const int* __restrict__ dst,
                             const float* __restrict__ xd,
                             float* __restrict__ t, int E) {
  int e = blockIdx.x * blockDim.x + threadIdx.x;
  if (e < E) atomicAdd(&t[dst[e]], xd[src[e]]);
}

// ---- h1 = relu(rank-1 + bias) --------------------------------------------
__global__ void gcn_h1(const float* __restrict__ dinv,
                       const float* __restrict__ t,
                       const float* __restrict__ xd,
                       const float* __restrict__ W1,
                       const float* __restrict__ b1,
                       float* __restrict__ h1, int N) {
  int i = blockIdx.x * blockDim.x + threadIdx.x;
  if (i < N) {
    float s = dinv[i] * (t[i] + xd[i]);
    float4* outp = (float4*)(h1 + i * 16);
#pragma unroll
    for (int q = 0; q < 4; ++q) {
      float4 v;
      float a0 = s * W1[4 * q + 0] + b1[4 * q + 0];
      float a1 = s * W1[4 * q + 1] + b1[4 * q + 1];
      float a2 = s * W1[4 * q + 2] + b1[4 * q + 2];
      float a3 = s * W1[4 * q + 3] + b1[4 * q + 3];
      v.x = a0 > 0.0f ? a0 : 0.0f;
      v.y = a1 > 0.0f ? a1 : 0.0f;
      v.z = a2 > 0.0f ? a2 : 0.0f;
      v.w = a3 > 0.0f ? a3 : 0.0f;
      outp[q] = v;
    }
  }
}

// ---- g = dinv * (h1 @ W2) via V_WMMA_F32_16X16X4_F32 ---------------------
// One wave per 16-row tile; B = W2 (16x2) zero-padded to 16 columns.
// A layout (16x4 f32): lane&15 = M; half = lane>>4 selects K {+0,+1} / {+2,+3}.
// B layout (4x16 f32): lane&15 = Ncol; half selects K rows {k0,k0+1}/{k0+2,k0+3}.
// D layout (16x16 f32): lanes 0-15 hold M=0..7 (VGPR j = row j) at N=lane;
//                       lanes 16-31 hold M=8..15 at N=lane-16.
// B padding uses clamped-index loads + arithmetic mask (no EXEC divergence,
// keeping EXEC all-ones around the WMMA chain with no save/restore churn).
__global__ __launch_bounds__(256) void gcn_h2_wmma(
    const float* __restrict__ h1, const float* __restrict__ dinv,
    const float* __restrict__ W2, float* __restrict__ g, int N, int ntiles) {
  int wave = threadIdx.x >> 5;
  int tile = blockIdx.x * 8 + wave;
  if (tile >= ntiles) return;                 // whole-wave guard: EXEC stays full
  int lane = threadIdx.x & 31;
  int col = lane & 15;
  int half = lane >> 4;
  int r0 = tile * 16;

  int arow = r0 + col;
  if (arow > N - 1) arow = N - 1;             // clamp pad rows (outputs unused)

  const int safec = (col < 2) ? col : 0;      // always-valid W2 column
  const float bmask = (col < 2) ? 1.0f : 0.0f;

  v8f acc = {};
#pragma unroll
  for (int k0 = 0; k0 < 16; k0 += 4) {
    int kk = k0 + half * 2;
    const float* ap = h1 + arow * 16 + kk;
    v2f a;
    a.x = ap[0];
    a.y = ap[1];
    v2f b;
    b.x = W2[kk * 2 + safec] * bmask;
    b.y = W2[(kk + 1) * 2 + safec] * bmask;
    acc = __builtin_amdgcn_wmma_f32_16x16x4_f32(false, a, false, b, (short)0,
                                                acc, false, false);
  }

  if (col < 2) {                              // only output columns 0,1 are real
#pragma unroll
    for (int j = 0; j < 8; ++j) {
      int row = r0 + half * 8 + j;
      if (row < N) g[row * 2 + col] = acc[j] * dinv[row];
    }
  }
}

// ---- layer-2 scatter: u[dst][0:2] += g[src][0:2] -------------------------
__global__ void gcn_scatter2(const int* __restrict__ src,
                             const int* __restrict__ dst,
                             const float* __restrict__ g,
                             float* __restrict__ u, int E) {
  int e = blockIdx.x * blockDim.x + threadIdx.x;
  if (e < E) {
    int s = src[e];
    int d = dst[e];
    float g0 = g[2 * s + 0];
    float g1 = g[2 * s + 1];
    atomicAdd(&u[2 * d + 0], g0);
    atomicAdd(&u[2 * d + 1], g1);
  }
}

// ---- bias + log_softmax ---------------------------------------------------
__global__ void gcn_out(const float* __restrict__ dinv,
                        const float* __restrict__ u,
                        const float* __restrict__ g,
                        const float* __restrict__ b2, float* __restrict__ out,
                        int N) {
  int i = blockIdx.x * blockDim.x + threadIdx.x;
  if (i < N) {
    float di = dinv[i];
    float o0 = di * (u[2 * i + 0] + g[2 * i + 0]) + b2[0];
    float o1 = di * (u[2 * i + 1] + g[2 * i + 1]) + b2[1];
    float m = fmaxf(o0, o1);
    float l = m + logf(expf(o0 - m) + expf(o1 - m));
    out[2 * i + 0] = o0 - l;
    out[2 * i + 1] = o1 - l;
  }
}

extern "C" void kernel_launch(void* const* d_in, const int* in_sizes, int n_in,
                              void* d_out, int out_size, void* d_ws,
                              size_t ws_size, hipStream_t stream) {
  const float* x = (const float*)d_in[0];
  const int* eidx = (const int*)d_in[1];   // [2, E] row-major int32
  const float* W1 = (const float*)d_in[2]; // [1,16]
  const float* b1 = (const float*)d_in[3]; // [16]
  const float* W2 = (const float*)d_in[4]; // [16,2]
  const float* b2 = (const float*)d_in[5]; // [2]
  float* out = (float*)d_out;

  const int N = in_sizes[0];
  const int E = in_sizes[1] / 2;
  const int* src = eidx;
  const int* dst = eidx + E;

  // workspace layout (floats)
  float* W = (float*)d_ws;
  float* dinv = W;           // [N]   (holds deg first, then dinv)
  float* xd = W + (size_t)N; // [N]
  float* t = W + 2 * (size_t)N;  // [N]
  float* h1 = W + 3 * (size_t)N; // [N*16]
  float* g = W + 19 * (size_t)N; // [N*2]
  float* u = W + 21 * (size_t)N; // [N*2]   total 23*N floats (~9.2 MB)

  const int TB = 256;
  const int gN = (N + TB - 1) / TB;
  const int gE = (E + TB - 1) / TB;
  const int ntiles = (N + 15) / 16;
  const int gT = (ntiles + 7) / 8; // 8 waves per block

  gcn_init<<<gN, TB, 0, stream>>>(dinv, t, u, N);
  gcn_degree<<<gE, TB, 0, stream>>>(dst, dinv, E);
  gcn_dinv<<<gN, TB, 0, stream>>>(x, dinv, xd, N);
  gcn_scatter1<<<gE, TB, 0, stream>>>(src, dst, xd, t, E);
  gcn_h1<<<gN, TB, 0, stream>>>(dinv, t, xd, W1, b1, h1, N);
  gcn_h2_wmma<<<gT, 256, 0, stream>>>(h1, dinv, W2, g, N, ntiles);
  gcn_scatter2<<<gE, TB, 0, stream>>>(src, dst, g, u, E);
  gcn_out<<<gN, TB, 0, stream>>>(dinv, u, g, b2, out, N);
}